// EmbeddingMapper_21801253995156
// MI455X (gfx1250) — compile-verified
//
#include <hip/hip_runtime.h>
#include <stdint.h>

// ---------------------------------------------------------------------------
// EmbeddingMapper (gather + OOV select), MI455X / gfx1250.
//
// Pure memory-bound op: ~52.5 MB total traffic -> ~2.3 us at 23.3 TB/s.
// Zero FLOPs, so no WMMA. The CDNA5-specific path used here is the async
// global<->LDS DMA (ASYNCcnt): gathered 16B row fragments are DMA'd
// HBM -> LDS -> HBM with no VGPR data round-trip.
//
// D4 (float4 chunks per embedding row) is a template constant so the
// token/chunk decomposition (e / D4, e % D4) lowers to mul-hi + shift
// instead of a runtime-division reciprocal sequence.
// ---------------------------------------------------------------------------

template <int D4>
__global__ __launch_bounds__(256) void embed_gather_async_kernel(
    const int* __restrict__ idx,     // [T] token indices (int32)
    const float* __restrict__ emb,   // [V, D]
    const float* __restrict__ oov,   // [D]
    float* __restrict__ out,         // [T, D]
    int vocab)                       // V
{
    // 4 slots per thread * 256 threads * 16 B = 16 KB LDS per block.
    __shared__ __align__(16) float lbuf[4 * 256 * 4];

    const unsigned tid   = threadIdx.x;
    const unsigned gsize = gridDim.x * 256u;          // total threads
    const unsigned e0    = blockIdx.x * 256u + tid;   // first float4 element

    constexpr unsigned row_bytes = (unsigned)D4 * 16u;  // bytes per row
    const uint64_t emb_b = (uint64_t)(uintptr_t)emb;
    const uint64_t oov_b = (uint64_t)(uintptr_t)oov;
    const uint64_t out_b = (uint64_t)(uintptr_t)out;

    unsigned lds_addr[4];
    uint64_t src[4];
    uint64_t dst[4];

#pragma unroll
    for (int j = 0; j < 4; ++j) {
        const unsigned e = e0 + (unsigned)j * gsize;  // float4 element id
        const unsigned t = e / (unsigned)D4;          // token (mulhi+shift)
        const unsigned c = e - t * (unsigned)D4;      // float4 chunk in row
        const int      w = idx[t];
        const uint64_t cb = (uint64_t)(c * 16u);
        src[j] = (w >= vocab) ? (oov_b + cb)
                              : (emb_b + (uint64_t)((unsigned)w) * row_bytes + cb);
        dst[j] = out_b + (uint64_t)e * 16u;
        // Low 32 bits of a generic pointer into LDS are the LDS byte address.
        lds_addr[j] = (unsigned)(uintptr_t)&lbuf[((unsigned)j * 256u + tid) * 4u];
    }

    // Batch 4 async gathers into LDS (2 KB in flight per wave).
#pragma unroll
    for (int j = 0; j < 4; ++j) {
        asm volatile("global_load_async_to_lds_b128 %0, %1, off"
                     :: "v"(lds_addr[j]), "v"(src[j])
                     : "memory");
    }
    // Wait for the LDS writes to land (ASYNCcnt == 0).
    asm volatile("s_wait_asynccnt 0x0" ::: "memory");
    // Stream the staged data straight from LDS to the output; these drain at
    // s_endpgm (implicit wait-idle), no explicit wait needed.
#pragma unroll
    for (int j = 0; j < 4; ++j) {
        asm volatile("global_store_async_from_lds_b128 %0, %1, off"
                     :: "v"(dst[j]), "v"(lds_addr[j])
                     : "memory");
    }
}

// Generic scalar tail / fallback (handles any size & the D % 4 != 0 case).
__global__ void embed_gather_tail_kernel(
    const int* __restrict__ idx,
    const float* __restrict__ emb,
    const float* __restrict__ oov,
    float* __restrict__ out,
    int start,      // first float element to process
    int total,      // total float elements
    int D,
    int vocab)
{
    int i = start + (int)(blockIdx.x * blockDim.x + threadIdx.x);
    if (i >= total) return;
    int t = i / D;
    int c = i - t * D;
    int w = idx[t];
    out[i] = (w >= vocab) ? oov[c] : emb[(size_t)w * (size_t)D + (size_t)c];
}

extern "C" void kernel_launch(void* const* d_in, const int* in_sizes, int n_in,
                              void* d_out, int out_size, void* d_ws, size_t ws_size,
                              hipStream_t stream) {
    const int*   idx = (const int*)d_in[0];    // word_indices [64,512] int32
    const float* emb = (const float*)d_in[1];  // embedding_matrix [V,D] f32
    const float* oov = (const float*)d_in[2];  // oov_embedding [1,D] f32
    float*       out = (float*)d_out;          // [64,512,D] f32

    const int T     = in_sizes[0];             // 32768 tokens
    const int D     = out_size / T;            // 200
    const int vocab = in_sizes[1] / D;         // 100000

    int covered_floats = 0;
    if (D == 200) {                            // fast path for the fixed shape
        const int n4       = out_size / 4;     // 1,638,400 float4
        const int per_grid = 256 * 4;          // float4 per block
        const int blocks   = n4 / per_grid;    // 1600 (exact fit)
        if (blocks > 0) {
            embed_gather_async_kernel<200 / 4><<<blocks, 256, 0, stream>>>(
                idx, emb, oov, out, vocab);
            covered_floats = blocks * per_grid * 4;
        }
    }

    const int rem = out_size - covered_floats;
    if (rem > 0) {
        const int tb = (rem + 255) / 256;
        embed_gather_tail_kernel<<<tb, 256, 0, stream>>>(
            idx, emb, oov, out, covered_floats, out_size, D, vocab);
    }
}